// TransModel_20761871909089
// MI455X (gfx1250) — compile-verified
//
#include <hip/hip_runtime.h>
#include <math.h>

typedef __attribute__((ext_vector_type(2))) float v2f;
typedef __attribute__((ext_vector_type(8))) float v8f;

#define SEQ   88
#define TT    64
#define HPAD  56     // 53 input features padded to multiple of 4
#define NGATE 48
#define HLn   12
#define NUc   24
#define NHh   4
#define HDd   6
#define KW    25
#define NPOS  (TT*SEQ)   // 5632

// padded weight sizes (floats)
#define WIH0P_SZ (96*56)
#define QKVWP_SZ (80*24)
#define PROJWP_SZ (32*24)

// ---------------------------------------------------------------------------
// Build cat = concat(features^T, emb[condition], mask) reshaped to (W*T, 56)
// row = w*64 + t ; cols 53..55 zero-padded.
// ---------------------------------------------------------------------------
__global__ void build_cat_kernel(const float* __restrict__ features,
                                 const int* __restrict__ condition,
                                 const float* __restrict__ mask,
                                 const float* __restrict__ emb,
                                 float* __restrict__ catb) {
  int idx = blockIdx.x * blockDim.x + threadIdx.x;
  if (idx >= NPOS * HPAD) return;
  int c   = idx % HPAD;
  int row = idx / HPAD;          // w*64 + t
  int w = row / TT, t = row % TT;
  float v = 0.f;
  if (c < 48)        v = features[(t * 48 + c) * SEQ + w];
  else if (c < 52)   v = emb[condition[t * SEQ + w] * 4 + (c - 48)];
  else if (c == 52)  v = mask[t * SEQ + w];
  catb[idx] = v;
}

// ---------------------------------------------------------------------------
// Zero-pad weight matrices into (N16 x Kpad) buffers so the WMMA GEMM inner
// loop needs no bounds checks at all.
// ---------------------------------------------------------------------------
__global__ void pad_weights_kernel(const float* __restrict__ wih0,   // 96x53
                                   const float* __restrict__ qkv_w,  // 72x24
                                   const float* __restrict__ proj_w, // 24x24
                                   float* __restrict__ wih0p,        // 96x56
                                   float* __restrict__ qkvwp,        // 80x24
                                   float* __restrict__ projwp) {     // 32x24
  int idx = blockIdx.x * blockDim.x + threadIdx.x;
  if (idx < WIH0P_SZ) {
    int n = idx / 56, k = idx % 56;
    wih0p[idx] = (k < 53) ? wih0[n * 53 + k] : 0.f;
  } else if (idx < WIH0P_SZ + QKVWP_SZ) {
    int r = idx - WIH0P_SZ;
    int n = r / 24, k = r % 24;
    qkvwp[r] = (n < 72) ? qkv_w[n * 24 + k] : 0.f;
  } else if (idx < WIH0P_SZ + QKVWP_SZ + PROJWP_SZ) {
    int r = idx - (WIH0P_SZ + QKVWP_SZ);
    int n = r / 24, k = r % 24;
    projwp[r] = (n < 24) ? proj_w[n * 24 + k] : 0.f;
  }
}

// ---------------------------------------------------------------------------
// fp32 WMMA GEMM: C(M x N) = A(M x kpad, lda=kpad) * Wp(N16 x kpad)^T + bias
// One wave32 per 16x16 C tile; V_WMMA_F32_16X16X4_F32; K fully unrolled.
// A zero-padded to lda; Wp zero-padded to N16 x kpad -> branch-free b64 loads.
// ---------------------------------------------------------------------------
template <int KITERS>
__global__ void gemm_wmma(const float* __restrict__ A, int lda,
                          const float* __restrict__ Wp, int kpad,
                          const float* __restrict__ bias0,
                          const float* __restrict__ bias1,
                          float* __restrict__ C, int N) {
  int lane = threadIdx.x;                 // 0..31
  int mt = blockIdx.x;
  int nt = blockIdx.y;
  int rA   = mt * 16 + (lane & 15);       // A row for this lane
  int colN = nt * 16 + (lane & 15);       // C column / Wp row for this lane
  int koff = (lane >> 4) * 2;             // lanes 0-15: K0,K1 ; 16-31: K2,K3

  const float* ap = A  + rA   * lda  + koff;
  const float* bp = Wp + colN * kpad + koff;

  v8f acc = {0.f, 0.f, 0.f, 0.f, 0.f, 0.f, 0.f, 0.f};
#pragma unroll
  for (int kk = 0; kk < KITERS; ++kk) {
    v2f a = *(const v2f*)(ap + kk * 4);
    v2f b = *(const v2f*)(bp + kk * 4);
    acc = __builtin_amdgcn_wmma_f32_16x16x4_f32(false, a, false, b,
                                                (short)0, acc, false, false);
  }
  if (colN < N) {
    float bv = bias0 ? bias0[colN] : 0.f;
    if (bias1) bv += bias1[colN];
    int r0 = mt * 16 + ((lane >> 4) ? 8 : 0);   // rows 0-7 / 8-15 per lane half
#pragma unroll
    for (int r = 0; r < 8; ++r)
      C[(r0 + r) * N + colN] = acc[r] + bv;
  }
}

// ---------------------------------------------------------------------------
// Bidirectional LSTM recurrence. One wave32 per (sequence, direction).
// xg holds precomputed x@Wih^T + b_ih + b_hh for both dirs: [seq*64+t][96].
// mode 0: write y[(s*64+t)*24 + d*12 + l]   (layer-0 -> layer-1 layout)
// mode 1: write y[(t*88+s)*24 + d*12 + l]   (attention (T,W,C) layout)
// ---------------------------------------------------------------------------
__global__ void lstm_rec(const float* __restrict__ xg,
                         const float* __restrict__ whh,   // [2][48][12]
                         float* __restrict__ y, int mode) {
  int s = blockIdx.x >> 1;
  int d = blockIdx.x & 1;
  int lane = threadIdx.x;
  __shared__ float h[HLn], c[HLn], g[NGATE];

  const float* wbase = whh + d * NGATE * HLn;
  float w0[HLn], w1[HLn];
#pragma unroll
  for (int k = 0; k < HLn; ++k) w0[k] = wbase[lane * HLn + k];
  if (lane < 16) {
#pragma unroll
    for (int k = 0; k < HLn; ++k) w1[k] = wbase[(32 + lane) * HLn + k];
  }
  if (lane < HLn) { h[lane] = 0.f; c[lane] = 0.f; }
  __syncthreads();

  for (int step = 0; step < TT; ++step) {
    int t = d ? (TT - 1 - step) : step;
    const float* xr = xg + (s * TT + t) * 96 + d * NGATE;
    float hloc[HLn];
#pragma unroll
    for (int k = 0; k < HLn; ++k) hloc[k] = h[k];
    float a0 = xr[lane];
#pragma unroll
    for (int k = 0; k < HLn; ++k) a0 += w0[k] * hloc[k];
    g[lane] = a0;
    if (lane < 16) {
      float a1 = xr[32 + lane];
#pragma unroll
      for (int k = 0; k < HLn; ++k) a1 += w1[k] * hloc[k];
      g[32 + lane] = a1;
    }
    __syncthreads();
    if (lane < HLn) {
      float ig = 1.f / (1.f + __expf(-g[lane]));
      float fg = 1.f / (1.f + __expf(-g[HLn + lane]));
      float gg = tanhf(g[2 * HLn + lane]);
      float og = 1.f / (1.f + __expf(-g[3 * HLn + lane]));
      float cn = fg * c[lane] + ig * gg;
      c[lane] = cn;
      float hn = og * tanhf(cn);
      h[lane] = hn;
      int oidx = mode ? ((t * SEQ + s) * NUc + d * HLn + lane)
                      : ((s * TT + t) * NUc + d * HLn + lane);
      y[oidx] = hn;
    }
    __syncthreads();
  }
}

// ---------------------------------------------------------------------------
// 2D neighborhood attention (K=25, 4 heads, head_dim 6) with online softmax.
// One thread per query; single pass over the 25x25 window; k/v rows loaded
// as float4 (b128) — row stride 288B and offsets 96B/192B are 16B-aligned.
// ---------------------------------------------------------------------------
__global__ void natten_kernel(const float* __restrict__ qkv,   // [5632][72]
                              const float* __restrict__ rpb,   // [4][49][49]
                              float* __restrict__ out) {       // [5632][24]
  int p = blockIdx.x * blockDim.x + threadIdx.x;
  if (p >= NPOS) return;
  int i = p / SEQ, j = p % SEQ;
  int ib = i - (KW / 2); ib = ib < 0 ? 0 : (ib > TT - KW ? TT - KW : ib);
  int jb = j - (KW / 2); jb = jb < 0 ? 0 : (jb > SEQ - KW ? SEQ - KW : jb);

  const float scale = 0.4082482904638630f;   // 1/sqrt(6)
  float q[NUc], m[NHh], l[NHh], acc[NUc];
  {
    const float4* qp4 = (const float4*)(qkv + p * 72);
#pragma unroll
    for (int v4 = 0; v4 < 6; ++v4) {
      float4 tq = qp4[v4];
      q[v4 * 4 + 0] = tq.x * scale; q[v4 * 4 + 1] = tq.y * scale;
      q[v4 * 4 + 2] = tq.z * scale; q[v4 * 4 + 3] = tq.w * scale;
    }
  }
#pragma unroll
  for (int n = 0; n < NHh; ++n) { m[n] = -1e30f; l[n] = 0.f; }
#pragma unroll
  for (int x = 0; x < NUc; ++x) acc[x] = 0.f;

  for (int a = 0; a < KW; ++a) {
    int ri = ib + a - i + (KW - 1);
    int rowbase = (ib + a) * SEQ;
    for (int cc = 0; cc < KW; ++cc) {
      int rj = jb + cc - j + (KW - 1);
      const float* rowp = qkv + (rowbase + jb + cc) * 72;
      float karr[NUc], varr[NUc];
      {
        const float4* kp4 = (const float4*)(rowp + NUc);
        const float4* vp4 = (const float4*)(rowp + 2 * NUc);
#pragma unroll
        for (int v4 = 0; v4 < 6; ++v4) {
          float4 tk = kp4[v4];
          karr[v4 * 4 + 0] = tk.x; karr[v4 * 4 + 1] = tk.y;
          karr[v4 * 4 + 2] = tk.z; karr[v4 * 4 + 3] = tk.w;
          float4 tv = vp4[v4];
          varr[v4 * 4 + 0] = tv.x; varr[v4 * 4 + 1] = tv.y;
          varr[v4 * 4 + 2] = tv.z; varr[v4 * 4 + 3] = tv.w;
        }
      }
#pragma unroll
      for (int n = 0; n < NHh; ++n) {
        float lg = rpb[(n * 49 + ri) * 49 + rj];
#pragma unroll
        for (int dd = 0; dd < HDd; ++dd) lg += q[n * HDd + dd] * karr[n * HDd + dd];
        float mo = m[n];
        float mn = fmaxf(mo, lg);
        float sc = __expf(mo - mn);
        float pw = __expf(lg - mn);
        l[n] = l[n] * sc + pw;
#pragma unroll
        for (int dd = 0; dd < HDd; ++dd)
          acc[n * HDd + dd] = acc[n * HDd + dd] * sc + pw * varr[n * HDd + dd];
        m[n] = mn;
      }
    }
  }
  float4* op4 = (float4*)(out + p * NUc);
#pragma unroll
  for (int n = 0; n < NHh; ++n) {
    float inv = 1.f / l[n];
#pragma unroll
    for (int dd = 0; dd < HDd; ++dd) acc[n * HDd + dd] *= inv;
  }
#pragma unroll
  for (int v4 = 0; v4 < 6; ++v4) {
    float4 tv;
    tv.x = acc[v4 * 4 + 0]; tv.y = acc[v4 * 4 + 1];
    tv.z = acc[v4 * 4 + 2]; tv.w = acc[v4 * 4 + 3];
    op4[v4] = tv;
  }
}

// ---------------------------------------------------------------------------
// Final 24 -> 5 projection.
// ---------------------------------------------------------------------------
__global__ void final_kernel(const float* __restrict__ x,
                             const float* __restrict__ ow,   // [5][24]
                             const float* __restrict__ ob,   // [5]
                             float* __restrict__ out) {
  int p = blockIdx.x * blockDim.x + threadIdx.x;
  if (p >= NPOS) return;
  float xr[NUc];
  const float4* xp4 = (const float4*)(x + p * NUc);
#pragma unroll
  for (int v4 = 0; v4 < 6; ++v4) {
    float4 t = xp4[v4];
    xr[v4 * 4 + 0] = t.x; xr[v4 * 4 + 1] = t.y;
    xr[v4 * 4 + 2] = t.z; xr[v4 * 4 + 3] = t.w;
  }
#pragma unroll
  for (int o = 0; o < 5; ++o) {
    float a = ob[o];
#pragma unroll
    for (int c = 0; c < NUc; ++c) a += xr[c] * ow[o * NUc + c];
    out[p * 5 + o] = a;
  }
}

// ---------------------------------------------------------------------------
extern "C" void kernel_launch(void* const* d_in, const int* in_sizes, int n_in,
                              void* d_out, int out_size, void* d_ws, size_t ws_size,
                              hipStream_t stream) {
  const float* features  = (const float*)d_in[0];
  const int*   condition = (const int*)  d_in[1];
  const float* mask      = (const float*)d_in[2];
  const float* emb       = (const float*)d_in[3];
  const float* wih0      = (const float*)d_in[4];   // (2,48,53) == 96x53
  const float* whh0      = (const float*)d_in[5];   // (2,48,12)
  const float* bih0      = (const float*)d_in[6];   // (2,48) == 96
  const float* bhh0      = (const float*)d_in[7];
  const float* wih1      = (const float*)d_in[8];   // 96x24 (already padded shape)
  const float* whh1      = (const float*)d_in[9];
  const float* bih1      = (const float*)d_in[10];
  const float* bhh1      = (const float*)d_in[11];
  const float* qkv_w     = (const float*)d_in[12];  // 72x24
  const float* qkv_b     = (const float*)d_in[13];
  const float* rpb       = (const float*)d_in[14];  // 4x49x49
  const float* proj_w    = (const float*)d_in[15];  // 24x24
  const float* proj_b    = (const float*)d_in[16];
  const float* out_w     = (const float*)d_in[17];  // 5x24
  const float* out_b     = (const float*)d_in[18];
  float* out = (float*)d_out;

  float* ws     = (float*)d_ws;
  float* catb   = ws;                     // 5632*56
  float* xg     = catb   + NPOS * HPAD;   // 5632*96
  float* y0     = xg     + NPOS * 96;     // 5632*24
  float* xA     = y0     + NPOS * NUc;    // 5632*24
  float* xB     = xA     + NPOS * NUc;    // 5632*24
  float* qkvb   = xB     + NPOS * NUc;    // 5632*72
  float* attnb  = qkvb   + NPOS * 72;     // 5632*24
  float* wih0p  = attnb  + NPOS * NUc;    // 96*56
  float* qkvwp  = wih0p  + WIH0P_SZ;      // 80*24
  float* projwp = qkvwp  + QKVWP_SZ;      // 32*24

  // 0) pad weights (branch-free GEMM inner loops)
  pad_weights_kernel<<<(WIH0P_SZ + QKVWP_SZ + PROJWP_SZ + 255) / 256, 256, 0,
                       stream>>>(wih0, qkv_w, proj_w, wih0p, qkvwp, projwp);
  // 1) cat input (zero-padded K)
  build_cat_kernel<<<(NPOS * HPAD + 255) / 256, 256, 0, stream>>>(
      features, condition, mask, emb, catb);
  // 2) layer-0 input projection for all timesteps/dirs (WMMA f32, K=56)
  gemm_wmma<14><<<dim3(NPOS / 16, 6), 32, 0, stream>>>(catb, HPAD, wih0p, 56,
                                                       bih0, bhh0, xg, 96);
  // 3) layer-0 recurrence (one wave per seq x dir)
  lstm_rec<<<SEQ * 2, 32, 0, stream>>>(xg, whh0, y0, 0);
  // 4) layer-1 input projection (WMMA f32, K=24)
  gemm_wmma<6><<<dim3(NPOS / 16, 6), 32, 0, stream>>>(y0, 24, wih1, 24,
                                                      bih1, bhh1, xg, 96);
  // 5) layer-1 recurrence, output directly in (T,W,C) attention layout
  lstm_rec<<<SEQ * 2, 32, 0, stream>>>(xg, whh1, xA, 1);
  // 6) two NATTEN rounds: QKV gemm -> fused attention -> proj gemm
  for (int r = 0; r < 2; ++r) {
    float* src = r ? xB : xA;
    float* dst = r ? xA : xB;
    gemm_wmma<6><<<dim3(NPOS / 16, 5), 32, 0, stream>>>(src, 24, qkvwp, 24,
                                                        qkv_b, nullptr, qkvb, 72);
    natten_kernel<<<(NPOS + 127) / 128, 128, 0, stream>>>(qkvb, rpb, attnb);
    gemm_wmma<6><<<dim3(NPOS / 16, 2), 32, 0, stream>>>(attnb, 24, projwp, 24,
                                                        proj_b, nullptr, dst, 24);
  }
  // 7) output head
  final_kernel<<<(NPOS + 255) / 256, 256, 0, stream>>>(xA, out_w, out_b, out);
}